// reversibleConvolution_9534827397300
// MI455X (gfx1250) — compile-verified
//
#include <hip/hip_runtime.h>

// reversibleConvolution on MI455X (gfx1250, wave32).
//
// Math: kernel is broadcast over output channels (setup uses randn(1,C,K)
// expanded to (C,C,K)), and N = L+K-1 makes the FFT conv an exact linear
// "same" conv:
//   y[b,t]     = sum_{c,k} x[b,c,t+15-k] * w[c,k],   w = kernel[o=0]
//   out[b,o,t] = y[b,t]  for every o
// Roofline: 134 MB read + 134 MB write = 268 MB -> ~11.5 us @ 23.3 TB/s;
// compute (after the 64x broadcast saving) is ~4.3 GFLOP of f32 GEMM,
// done on the matrix pipe as U[tau,k] = sum_c x[b,c,tau]*w[c,k] via
// V_WMMA_F32_16X16X4_F32, then the diagonal gather y[t] = sum_k U[t+15-k,k].
// fp32 WMMA keeps accuracy on par with the f32 FFT reference.

typedef float v2f __attribute__((ext_vector_type(2)));
typedef float v8f __attribute__((ext_vector_type(8)));

#define B_    128
#define C_    64
#define L_    4096
#define K_    31
#define T_    64      // output t-values per workgroup
#define NTILE 6       // tau-tiles of 16 per workgroup (96 tau = 64 + 2*16 halo)

__global__ __launch_bounds__(64)
void reversibleConvolution_wmma_kernel(const float* __restrict__ x,
                                       const float* __restrict__ ker,
                                       float* __restrict__ out)
{
    __shared__ float wlds[C_ * 33];               // w[c,k], padded stride 33
    __shared__ float ulds[(NTILE * 16) * 33];     // U[tau-row, k], padded stride 33
    __shared__ __align__(16) float ylds[T_];      // y staging for vectorized stores

    const int tid  = threadIdx.x;
    const int lane = tid & 31;
    const int wave = tid >> 5;
    const int hi   = lane >> 4;     // half-wave select (K-pair 0/1 vs 2/3)
    const int lo   = lane & 15;

    const int b    = blockIdx.y;
    const int t0   = blockIdx.x * T_;
    const int tau0 = t0 - 16;       // tile-aligned halo start

    // Stage broadcast kernel slice o==0 into LDS: wlds[c*33+k], k=31 -> 0 pad.
    for (int idx = tid; idx < C_ * 32; idx += 64) {
        const int c = idx >> 5, k = idx & 31;
        wlds[c * 33 + k] = (k < K_) ? ker[c * K_ + k] : 0.0f;
    }
    __syncthreads();

    const float* xb = x + (size_t)b * C_ * L_;

    // Each wave produces 3 of the 6 16-tau tiles of U[tau, 0..31].
    #pragma unroll
    for (int tt = 0; tt < NTILE / 2; ++tt) {
        const int tile = wave + 2 * tt;
        const int tau  = tau0 + tile * 16 + lo;
        // Unconditional loads: clamp the address, zero via multiply mask.
        const int   tauc = (tau < 0) ? 0 : ((tau > L_ - 1) ? L_ - 1 : tau);
        const float msk  = (tau >= 0 && tau < L_) ? 1.0f : 0.0f;
        const float* xt  = xb + tauc;

        v8f acc0 = {0.f,0.f,0.f,0.f,0.f,0.f,0.f,0.f};  // taps 0..15
        v8f acc1 = {0.f,0.f,0.f,0.f,0.f,0.f,0.f,0.f};  // taps 16..31

        #pragma unroll
        for (int cc = 0; cc < 16; ++cc) {              // C reduction, 4 at a time
            const int c = cc * 4 + 2 * hi;
            // A (16x4 f32): lane L<16 holds {K=0,K=1}, lane L>=16 holds {K=2,K=3}
            v2f a;
            a.x = msk * xt[(size_t)c       * L_];
            a.y = msk * xt[(size_t)(c + 1) * L_];
            // B (4x16 f32): mirrored layout, column = lo
            v2f b0, b1;
            b0.x = wlds[c       * 33 + lo];
            b0.y = wlds[(c + 1) * 33 + lo];
            b1.x = wlds[c       * 33 + 16 + lo];
            b1.y = wlds[(c + 1) * 33 + 16 + lo];

            acc0 = __builtin_amdgcn_wmma_f32_16x16x4_f32(
                       false, a, false, b0, (short)0, acc0, false, false);
            acc1 = __builtin_amdgcn_wmma_f32_16x16x4_f32(
                       false, a, false, b1, (short)0, acc1, false, false);
        }

        // D 16x16 layout: VGPR r = row r (lanes 0-15) / row r+8 (lanes 16-31)
        const int rbase = tile * 16 + hi * 8;
        #pragma unroll
        for (int r = 0; r < 8; ++r) {
            ulds[(rbase + r) * 33 + lo]      = acc0[r];
            ulds[(rbase + r) * 33 + 16 + lo] = acc1[r];
        }
    }
    __syncthreads();

    // Diagonal gather: y[t] = sum_{k=0..30} U[t+15-k, k]
    // row index = (t0+tid+15-k) - tau0 = tid + 31 - k  (stride-33 rows)
    float y = 0.0f;
    #pragma unroll
    for (int k = 0; k < K_; ++k)
        y += ulds[(tid + 31 - k) * 33 + k];

    ylds[tid] = y;
    __syncthreads();

    // Broadcast to all 64 output channels with float4 stores:
    // each iteration, 16-lane groups write one contiguous 256B channel row.
    float* ob = out + (size_t)b * C_ * L_ + t0;
    const float4* y4 = (const float4*)ylds;
    #pragma unroll
    for (int i = 0; i < (C_ * (T_ / 4)) / 64; ++i) {   // 16 iterations
        const int idx = i * 64 + tid;
        const int o   = idx >> 4;        // output channel
        const int t4  = idx & 15;        // float4 slot within the 64-t row
        ((float4*)(ob + (size_t)o * L_))[t4] = y4[t4];
    }
}

extern "C" void kernel_launch(void* const* d_in, const int* in_sizes, int n_in,
                              void* d_out, int out_size, void* d_ws, size_t ws_size,
                              hipStream_t stream) {
    const float* x   = (const float*)d_in[0];   // [128, 64, 4096] f32
    const float* ker = (const float*)d_in[1];   // [64, 64, 31] f32 (broadcast over dim 0)
    float*       out = (float*)d_out;           // [128, 64, 4096] f32

    dim3 grid(L_ / T_, B_);   // (64, 128)
    dim3 block(T_);           // 64 threads = 2 waves (wave32)
    reversibleConvolution_wmma_kernel<<<grid, block, 0, stream>>>(x, ker, out);
}